// DecoderTransformer_74328704024699
// MI455X (gfx1250) — compile-verified
//
#include <hip/hip_runtime.h>

// ---------------------------------------------------------------------------
// Decoder transformer forward, MI455X (gfx1250), all-f32 via V_WMMA_F32_16X16X4_F32.
// Memory-bound on the 524MB logits store; all weights/activations are L2-resident,
// so full-precision f32 WMMA is the right precision (no speed win from f16/fp8).
// ---------------------------------------------------------------------------

typedef __attribute__((ext_vector_type(2))) float v2f;
typedef __attribute__((ext_vector_type(8))) float v8f;

#define TT     1024
#define CEMB   64
#define NH     4
#define HS     16
#define DFF    256
#define NVOCAB 32000
#define MROWS  4096   // B*T

__device__ __forceinline__ v8f wmma4(v2f a, v2f b, v8f c) {
  // D = A(16x4,f32) * B(4x16,f32) + C(16x16,f32)
  return __builtin_amdgcn_wmma_f32_16x16x4_f32(false, a, false, b, (short)0, c,
                                               false, false);
}

// ---------------------------------------------------------------------------
// x = tok_emb[idx] + pos_emb[:T]
// ---------------------------------------------------------------------------
__global__ __launch_bounds__(256) void embed_k(const int* __restrict__ idx,
                                               const float* __restrict__ tok,
                                               const float* __restrict__ pos,
                                               float* __restrict__ x) {
  const int i = blockIdx.x * blockDim.x + threadIdx.x;
  if (i >= MROWS * CEMB) return;
  const int row = i >> 6;         // token row in [0,4096)
  const int c   = i & 63;
  const int t   = row & (TT - 1); // position within sequence
  x[i] = tok[(size_t)idx[row] * CEMB + c] + pos[t * CEMB + c];
}

// ---------------------------------------------------------------------------
// LayerNorm over the 64-wide channel dim, one thread per row.
// ---------------------------------------------------------------------------
__global__ __launch_bounds__(256) void lnorm_k(const float* __restrict__ x,
                                               const float* __restrict__ g,
                                               const float* __restrict__ bt,
                                               float* __restrict__ out) {
  const int row = blockIdx.x * blockDim.x + threadIdx.x;
  if (row >= MROWS) return;
  const float* xr = x + (size_t)row * CEMB;
  float buf[CEMB];
  float mean = 0.f;
#pragma unroll
  for (int c = 0; c < CEMB; ++c) { buf[c] = xr[c]; mean += buf[c]; }
  mean *= (1.f / CEMB);
  float var = 0.f;
#pragma unroll
  for (int c = 0; c < CEMB; ++c) { float d = buf[c] - mean; var += d * d; }
  var *= (1.f / CEMB);
  const float inv = rsqrtf(var + 1e-5f);
  float* orow = out + (size_t)row * CEMB;
#pragma unroll
  for (int c = 0; c < CEMB; ++c) orow[c] = (buf[c] - mean) * inv * g[c] + bt[c];
}

// ---------------------------------------------------------------------------
// Generic tiled GEMM: C[M,N] = A[M,K] * W[K,N] (+bias) (+resid) (relu?)
// 8 waves / block; one 16x16 tile per wave via K/4 chained wmma ops.
// ---------------------------------------------------------------------------
template <bool RELU, bool RESID, bool BIAS>
__global__ __launch_bounds__(256) void gemm16_k(const float* __restrict__ A,
                                                const float* __restrict__ W,
                                                const float* __restrict__ bias,
                                                const float* __restrict__ resid,
                                                float* __restrict__ Cout,
                                                int M, int N, int K) {
  const int wave = threadIdx.x >> 5;
  const int lane = threadIdx.x & 31;
  const int hi = lane >> 4, lo = lane & 15;
  const int ntile = N >> 4;
  const long tile = (long)blockIdx.x * 8 + wave;
  if (tile >= (long)(M >> 4) * ntile) return;   // whole-wave exit: EXEC stays full
  const int mt = (int)(tile / ntile);
  const int nt = (int)(tile % ntile);
  const float* Arow = A + (size_t)(mt * 16 + lo) * K;  // A: m = lane%16
  const float* Wcol = W + nt * 16 + lo;                // B: n = lane%16
  v8f acc = {};
  for (int kk = 0; kk < K; kk += 4) {
    const int ka = kk + 2 * hi;                        // k = r + 2*(lane/16)
    v2f a, b;
    a.x = Arow[ka];
    a.y = Arow[ka + 1];
    b.x = Wcol[(size_t)ka * N];
    b.y = Wcol[(size_t)(ka + 1) * N];
    acc = wmma4(a, b, acc);
  }
#pragma unroll
  for (int r = 0; r < 8; ++r) {                        // C: row = r + 8*(lane/16)
    const int row = mt * 16 + r + 8 * hi;
    const int col = nt * 16 + lo;
    float v = acc[r];
    if (BIAS)  v += bias[col];
    if (RESID) v += resid[(size_t)row * N + col];
    if (RELU)  v = fmaxf(v, 0.f);
    Cout[(size_t)row * N + col] = v;
  }
}

// ---------------------------------------------------------------------------
// QKV projection: per head h, [4096,64] x Wq[l,h][64,16] -> q[:, h*16:(h+1)*16]
// blockIdx = (m-tile group of 8, head, {q,k,v}); 8 waves, one m-tile each.
// ---------------------------------------------------------------------------
__global__ __launch_bounds__(256) void qkv16_k(const float* __restrict__ xn,
                                               const float* __restrict__ Wq,
                                               const float* __restrict__ Wk,
                                               const float* __restrict__ Wv,
                                               float* __restrict__ qb,
                                               float* __restrict__ kb,
                                               float* __restrict__ vb, int layer) {
  const int wave = threadIdx.x >> 5;
  const int lane = threadIdx.x & 31;
  const int hi = lane >> 4, lo = lane & 15;
  const int mt = blockIdx.x * 8 + wave;   // 0..255
  const int h  = blockIdx.y;
  const float* W;
  float* out;
  if (blockIdx.z == 0)      { W = Wq; out = qb; }
  else if (blockIdx.z == 1) { W = Wk; out = kb; }
  else                      { W = Wv; out = vb; }
  W += (size_t)(layer * NH + h) * (CEMB * HS);   // contiguous 64x16 block
  const float* Arow = xn + (size_t)(mt * 16 + lo) * CEMB;
  v8f acc = {};
  for (int kk = 0; kk < CEMB; kk += 4) {
    const int ka = kk + 2 * hi;
    v2f a, b;
    a.x = Arow[ka];
    a.y = Arow[ka + 1];
    b.x = W[ka * HS + lo];
    b.y = W[(ka + 1) * HS + lo];
    acc = wmma4(a, b, acc);
  }
#pragma unroll
  for (int r = 0; r < 8; ++r) {
    const int row = mt * 16 + r + 8 * hi;
    out[(size_t)row * CEMB + h * HS + lo] = acc[r];   // [B,T,H,16] concat layout
  }
}

// ---------------------------------------------------------------------------
// Causal flash attention, one wave per (q-tile, head, batch).
// Scores QK^T and PV both on WMMA; online softmax with 16-lane shfl_xor
// row reductions (row groups match the WMMA C layout exactly).
// ---------------------------------------------------------------------------
__global__ __launch_bounds__(32) void attn16_k(const float* __restrict__ qb,
                                               const float* __restrict__ kb,
                                               const float* __restrict__ vb,
                                               float* __restrict__ ab) {
  __shared__ float pshm[16 * 16];
  const int lane = threadIdx.x & 31;
  const int hi = lane >> 4, lo = lane & 15;
  const int qt = blockIdx.x, h = blockIdx.y, b = blockIdx.z;
  const int rowbase = b * TT;
  const int qbase = qt * 16;
  const float scale = 0.125f;  // 1/sqrt(N_EMBD) per reference

  // Preload Q A-fragments (M=16 queries, K=16 head dims).
  v2f qa[4];
#pragma unroll
  for (int c = 0; c < 4; ++c) {
    const int ka = 4 * c + 2 * hi;
    const float* qr = qb + (size_t)(rowbase + qbase + lo) * CEMB + h * HS;
    qa[c].x = qr[ka];
    qa[c].y = qr[ka + 1];
  }

  float mrun[8], lrun[8];
  v8f o = {};
#pragma unroll
  for (int r = 0; r < 8; ++r) { mrun[r] = -1e30f; lrun[r] = 0.f; }

  for (int st = 0; st <= qt; ++st) {
    // S = Q * K^T : B-frag element = K[s=lane%16][d = r+2*(lane/16)]
    v8f s = {};
#pragma unroll
    for (int c = 0; c < 4; ++c) {
      const int ka = 4 * c + 2 * hi;
      const float* kr = kb + (size_t)(rowbase + st * 16 + lo) * CEMB + h * HS;
      v2f bf;
      bf.x = kr[ka];
      bf.y = kr[ka + 1];
      s = wmma4(qa[c], bf, s);
    }
    // Online softmax update per row group.
#pragma unroll
    for (int r = 0; r < 8; ++r) {
      const int row = r + 8 * hi;            // query row within tile
      float sv = s[r] * scale;
      if (st == qt && lo > row) sv = -1e30f; // causal mask on diagonal tile
      float rmax = sv;
      for (int off = 1; off < 16; off <<= 1)
        rmax = fmaxf(rmax, __shfl_xor(rmax, off));
      const float mnew = fmaxf(mrun[r], rmax);
      const float p = __expf(sv - mnew);
      float rsum = p;
      for (int off = 1; off < 16; off <<= 1) rsum += __shfl_xor(rsum, off);
      const float alpha = __expf(mrun[r] - mnew);
      lrun[r] = lrun[r] * alpha + rsum;
      mrun[r] = mnew;
      o[r] *= alpha;
      pshm[row * 16 + lo] = p;               // stage P (C-layout -> LDS)
    }
    __syncthreads();
    // O += P * V : A-frag from LDS, B-frag = V[s][d] from global.
#pragma unroll
    for (int c = 0; c < 4; ++c) {
      const int ka = 4 * c + 2 * hi;
      v2f a, bf;
      a.x = pshm[lo * 16 + ka];
      a.y = pshm[lo * 16 + ka + 1];
      bf.x = vb[(size_t)(rowbase + st * 16 + ka) * CEMB + h * HS + lo];
      bf.y = vb[(size_t)(rowbase + st * 16 + ka + 1) * CEMB + h * HS + lo];
      o = wmma4(a, bf, o);
    }
    __syncthreads();
  }
#pragma unroll
  for (int r = 0; r < 8; ++r) {
    const int row = r + 8 * hi;
    ab[(size_t)(rowbase + qbase + row) * CEMB + h * HS + lo] = o[r] / lrun[r];
  }
}

// ---------------------------------------------------------------------------
// Host orchestration (all launches on `stream`, graph-capture safe).
// ---------------------------------------------------------------------------
extern "C" void kernel_launch(void* const* d_in, const int* in_sizes, int n_in,
                              void* d_out, int out_size, void* d_ws, size_t ws_size,
                              hipStream_t stream) {
  (void)in_sizes; (void)n_in; (void)out_size; (void)ws_size;
  const int* idx   = (const int*)d_in[0];
  const float* tok = (const float*)d_in[1];
  const float* pos = (const float*)d_in[2];
  const float* Wq  = (const float*)d_in[3];
  const float* Wk  = (const float*)d_in[4];
  const float* Wv  = (const float*)d_in[5];
  const float* Wo  = (const float*)d_in[6];
  const float* bo  = (const float*)d_in[7];
  const float* W1  = (const float*)d_in[8];
  const float* b1  = (const float*)d_in[9];
  const float* W2  = (const float*)d_in[10];
  const float* b2  = (const float*)d_in[11];
  const float* g1  = (const float*)d_in[12];
  const float* be1 = (const float*)d_in[13];
  const float* g2  = (const float*)d_in[14];
  const float* be2 = (const float*)d_in[15];
  const float* Wlm = (const float*)d_in[16];
  const float* blm = (const float*)d_in[17];
  float* out = (float*)d_out;

  // Workspace layout (floats): ~10.5 MB total.
  float* ws = (float*)d_ws;
  float* x  = ws;                 // [4096,64] residual stream
  float* xn = x  + MROWS * CEMB;  // [4096,64] LN output
  float* qb = xn + MROWS * CEMB;  // [B,T,H,16]
  float* kb = qb + MROWS * CEMB;
  float* vb = kb + MROWS * CEMB;
  float* ab = vb + MROWS * CEMB;  // attention output (head-concat)
  float* hb = ab + MROWS * CEMB;  // [4096,256] FFN hidden

  embed_k<<<(MROWS * CEMB + 255) / 256, 256, 0, stream>>>(idx, tok, pos, x);

  for (int l = 0; l < 4; ++l) {
    lnorm_k<<<16, 256, 0, stream>>>(x, g1 + l * CEMB, be1 + l * CEMB, xn);
    qkv16_k<<<dim3(32, 4, 3), 256, 0, stream>>>(xn, Wq, Wk, Wv, qb, kb, vb, l);
    attn16_k<<<dim3(64, 4, 4), 32, 0, stream>>>(qb, kb, vb, ab);
    // x = x + ab @ Wo[l] + bo[l]      (tiles = 256*4 = 1024 -> 128 blocks)
    gemm16_k<false, true, true><<<128, 256, 0, stream>>>(
        ab, Wo + (size_t)l * CEMB * CEMB, bo + l * CEMB, x, x, MROWS, CEMB, CEMB);
    lnorm_k<<<16, 256, 0, stream>>>(x, g2 + l * CEMB, be2 + l * CEMB, xn);
    // hb = relu(xn @ W1[l] + b1[l])   (tiles = 256*16 = 4096 -> 512 blocks)
    gemm16_k<true, false, true><<<512, 256, 0, stream>>>(
        xn, W1 + (size_t)l * CEMB * DFF, b1 + l * DFF, nullptr, hb, MROWS, DFF, CEMB);
    // x = x + hb @ W2[l] + b2[l]
    gemm16_k<false, true, true><<<128, 256, 0, stream>>>(
        hb, W2 + (size_t)l * DFF * CEMB, b2 + l * CEMB, x, x, MROWS, CEMB, DFF);
  }

  // Logits: [4096,64] @ [64,32000] + blm  (tiles = 256*2000 -> 64000 blocks)
  gemm16_k<false, false, true><<<64000, 256, 0, stream>>>(
      x, Wlm, blm, nullptr, out, MROWS, NVOCAB, CEMB);
}